// SigGNNExpert_20538533609910
// MI455X (gfx1250) — compile-verified
//
#include <hip/hip_runtime.h>
#include <math.h>

// ---------------------------------------------------------------------------
// CDNA5 (gfx1250) implementation of SigGNNExpert (GATv2 stack), fp32.
// Dense GEMMs run on the matrix pipe via V_WMMA_F32_16X16X4_F32 with
// LDS-staged 64x64 block tiles (coalesced global -> LDS -> fragments).
// ---------------------------------------------------------------------------

typedef __attribute__((ext_vector_type(2))) float v2f;
typedef __attribute__((ext_vector_type(8))) float v8f;

// ---- order-preserving float <-> uint encoding for atomic max on floats ----
__device__ __forceinline__ unsigned ford(float f) {
  unsigned u = __float_as_uint(f);
  return (u & 0x80000000u) ? ~u : (u | 0x80000000u);
}
__device__ __forceinline__ float orddec(unsigned u) {
  return (u & 0x80000000u) ? __uint_as_float(u ^ 0x80000000u)
                           : __uint_as_float(~u);
}
#define ORD_NEG_INF 0x007FFFFFu   // ford(-inf)

// ---------------------------------------------------------------------------
// Block GEMM: C[M,N] = A[M,K] @ W[K,N] (+ bias[N])
//   block tile 64x64, 8 waves: wave = (colhalf ch in {0,1}) x (rowtile rt in 0..3)
//   each wave: 16 rows x 32 cols = two 16x16 WMMA accumulators.
//   K chunked by 16; A (64x16) and W^T (64x16) staged in LDS, padded stride 20
//   words -> 16B-aligned float4 stores, conflict-free b64 fragment reads.
// Fragment layouts (V_WMMA_F32_16X16X4_F32):
//   A 16x4 : lane l -> A[l%16][k + 2*(l/16) + {0,1}]
//   B 4x16 : lane l -> W[k + 2*(l/16) + {0,1}][col0 + l%16]
//   C 16x16: VGPR r -> row r (lanes 0-15) / row r+8 (lanes 16-31), col l%16
// Requires: N % 64 == 0, K % 16 == 0 (M may be ragged).
// ---------------------------------------------------------------------------
#define BM 64
#define BN 64
#define KC 16
#define LDP 20   // padded LDS row stride in words

__global__ void gemm_wmma_f32_kernel(const float* __restrict__ A,
                                     const float* __restrict__ W,
                                     const float* __restrict__ bias,
                                     float* __restrict__ C,
                                     int M, int K, int N) {
  __shared__ float As[BM * LDP];   // As[row][kk]
  __shared__ float Wt[BN * LDP];   // Wt[col][kk]  (transposed W tile)

  int tid   = threadIdx.x;
  int wavei = tid >> 5;
  int lane  = tid & 31;
  int lmod  = lane & 15;
  int lhi   = lane >> 4;            // K phase: 0 -> k0/k1, 1 -> k2/k3

  int nbt  = N / BN;
  int mb   = blockIdx.x / nbt;
  int nb   = blockIdx.x - mb * nbt;
  int row0 = mb * BM, col0 = nb * BN;

  int rt = wavei & 3;               // 16-row tile within block
  int ch = wavei >> 2;              // 32-col half within block
  int wrow = rt * 16 + lmod;        // this lane's A row (within block)
  int wc0  = ch * 32;

  // staging indices: A as 256 float4 (thread t -> row t/4, k (t%4)*4)
  int arow = tid >> 2;
  int akk  = (tid & 3) * 4;
  int agrow = row0 + arow;
  if (agrow > M - 1) agrow = M - 1;           // clamp ragged tail (no early exit)

  v8f acc0 = {}, acc1 = {};

  for (int k0 = 0; k0 < K; k0 += KC) {
    // ---- global -> regs (coalesced) ----
    const float* ap = A + (size_t)agrow * K + k0 + akk;
    float4 av = *(const float4*)ap;
    float wv[4];
    #pragma unroll
    for (int i = 0; i < 4; ++i) {
      int idx = tid + i * 256;
      int kk = idx >> 6, c = idx & 63;
      wv[i] = W[(size_t)(k0 + kk) * N + col0 + c];
    }
    __syncthreads();                           // previous chunk fully consumed
    // ---- regs -> LDS ----
    *(float4*)&As[arow * LDP + akk] = av;
    #pragma unroll
    for (int i = 0; i < 4; ++i) {
      int idx = tid + i * 256;
      int kk = idx >> 6, c = idx & 63;
      Wt[c * LDP + kk] = wv[i];
    }
    __syncthreads();

    if (k0 + KC < K) __builtin_prefetch(ap + KC, 0, 0);  // next A chunk

    // ---- compute: 4 k-steps x 2 col tiles ----
    #pragma unroll
    for (int kk = 0; kk < KC; kk += 4) {
      int kf = kk + 2 * lhi;
      v2f a  = *(const v2f*)&As[wrow * LDP + kf];
      v2f b0 = *(const v2f*)&Wt[(wc0 + lmod) * LDP + kf];
      v2f b1 = *(const v2f*)&Wt[(wc0 + 16 + lmod) * LDP + kf];
      acc0 = __builtin_amdgcn_wmma_f32_16x16x4_f32(false, a, false, b0,
                                                   (short)0, acc0, false, false);
      acc1 = __builtin_amdgcn_wmma_f32_16x16x4_f32(false, a, false, b1,
                                                   (short)0, acc1, false, false);
    }
  }

  // ---- epilogue ----
  int ccol = col0 + wc0 + lmod;
  float bv0 = bias ? bias[ccol] : 0.0f;
  float bv1 = bias ? bias[ccol + 16] : 0.0f;
  int rbase = row0 + rt * 16 + lhi * 8;
  #pragma unroll
  for (int r = 0; r < 8; ++r) {
    int row = rbase + r;
    if (row < M) {
      C[(size_t)row * N + ccol]      = acc0[r] + bv0;
      C[(size_t)row * N + ccol + 16] = acc1[r] + bv1;
    }
  }
}

// ---------------------------------------------------------------------------
__global__ void fill_u32_kernel(unsigned* __restrict__ p, unsigned v, int n) {
  int i = blockIdx.x * blockDim.x + threadIdx.x;
  if (i < n) p[i] = v;
}

// tanh-approximate GELU (jax.nn.gelu default), in place
__global__ void gelu_kernel(float* __restrict__ x, int n) {
  int i = blockIdx.x * blockDim.x + threadIdx.x;
  if (i < n) {
    float v = x[i];
    float c = 0.7978845608028654f * (v + 0.044715f * v * v * v);
    x[i] = 0.5f * v * (1.0f + tanhf(c));
  }
}

// ---------------------------------------------------------------------------
// Row LayerNorm over 256 channels; optional residual add (Agg) and bias.
// One wave32 per row, 8 channels/lane, __shfl_xor tree reductions.
// ---------------------------------------------------------------------------
__global__ void resid_ln_kernel(const float* __restrict__ Hin,
                                const float* __restrict__ Agg,
                                const float* __restrict__ bias,
                                const float* __restrict__ g,
                                const float* __restrict__ beta,
                                float* __restrict__ Hout, int rows) {
  int row  = blockIdx.x * (blockDim.x >> 5) + (threadIdx.x >> 5);
  int lane = threadIdx.x & 31;
  if (row >= rows) return;
  size_t base = (size_t)row * 256 + lane * 8;
  float v[8];
  float s = 0.0f;
  #pragma unroll
  for (int i = 0; i < 8; ++i) {
    float t = Hin[base + i];
    if (Agg)  t += Agg[base + i];
    if (bias) t += bias[lane * 8 + i];
    v[i] = t; s += t;
  }
  #pragma unroll
  for (int m = 16; m >= 1; m >>= 1) s += __shfl_xor(s, m);
  float mu = s * (1.0f / 256.0f);
  float vs = 0.0f;
  #pragma unroll
  for (int i = 0; i < 8; ++i) { float d = v[i] - mu; vs += d * d; }
  #pragma unroll
  for (int m = 16; m >= 1; m >>= 1) vs += __shfl_xor(vs, m);
  float inv = rsqrtf(vs * (1.0f / 256.0f) + 1e-5f);
  #pragma unroll
  for (int i = 0; i < 8; ++i)
    Hout[base + i] = (v[i] - mu) * inv * g[lane * 8 + i] + beta[lane * 8 + i];
}

// ---------------------------------------------------------------------------
// Pass A: per-edge attention logits. One wave per edge; lane covers 8
// consecutive channels; head = lane/4. Partial dot with att reduced across
// the 4 lanes of each head; lane h*4 stores logit + atomic segment-max.
// ---------------------------------------------------------------------------
__global__ void edge_logits_kernel(const float* __restrict__ xl,
                                   const float* __restrict__ xr,
                                   const float* __restrict__ eproj,
                                   const int* __restrict__ ei,
                                   const float* __restrict__ att,
                                   float* __restrict__ logits,
                                   unsigned* __restrict__ nmax, int E) {
  int eid  = blockIdx.x * (blockDim.x >> 5) + (threadIdx.x >> 5);
  int lane = threadIdx.x & 31;
  if (eid >= E) return;
  int src = ei[eid], dst = ei[E + eid];
  const float* pj = xl    + (size_t)src * 256 + lane * 8;
  const float* pr = xr    + (size_t)dst * 256 + lane * 8;
  const float* pe = eproj + (size_t)eid * 256 + lane * 8;
  const float* pa = att + lane * 8;   // att[H=8,C=32] is channel-contiguous
  float p = 0.0f;
  #pragma unroll
  for (int i = 0; i < 8; ++i) {
    float s = pj[i] + pr[i] + pe[i];
    s = (s >= 0.0f) ? s : 0.2f * s;   // leaky_relu(0.2)
    p += s * pa[i];
  }
  p += __shfl_xor(p, 1);
  p += __shfl_xor(p, 2);
  if ((lane & 3) == 0) {
    int h = lane >> 2;
    logits[(size_t)eid * 8 + h] = p;
    atomicMax(&nmax[(size_t)dst * 8 + h], ford(p));
  }
}

// Pass B: numerators exp(logit - max[dst]) + segment sums
__global__ void edge_exp_kernel(const int* __restrict__ ei,
                                const unsigned* __restrict__ nmax,
                                float* __restrict__ logits,
                                float* __restrict__ nsum, int E) {
  int idx = blockIdx.x * blockDim.x + threadIdx.x;
  if (idx >= E * 8) return;
  int eid = idx >> 3, h = idx & 7;
  int dst = ei[E + eid];
  float m = orddec(nmax[(size_t)dst * 8 + h]);
  if (!__builtin_isfinite(m)) m = 0.0f;   // matches reference isfinite guard
  float w = __expf(logits[idx] - m);
  logits[idx] = w;
  atomicAdd(&nsum[(size_t)dst * 8 + h], w);
}

// Pass C: agg[dst] += xl[src] * alpha  (per head)
__global__ void edge_scatter_kernel(const float* __restrict__ xl,
                                    const int* __restrict__ ei,
                                    const float* __restrict__ logits,
                                    const float* __restrict__ nsum,
                                    float* __restrict__ agg, int E) {
  int eid  = blockIdx.x * (blockDim.x >> 5) + (threadIdx.x >> 5);
  int lane = threadIdx.x & 31;
  if (eid >= E) return;
  int src = ei[eid], dst = ei[E + eid];
  int h = lane >> 2;
  float alpha = logits[(size_t)eid * 8 + h] /
                (nsum[(size_t)dst * 8 + h] + 1e-16f);
  const float* pj = xl  + (size_t)src * 256 + lane * 8;
  float*       pd = agg + (size_t)dst * 256 + lane * 8;
  #pragma unroll
  for (int i = 0; i < 8; ++i) atomicAdd(&pd[i], pj[i] * alpha);
}

// ---------------------------------------------------------------------------
extern "C" void kernel_launch(void* const* d_in, const int* in_sizes, int n_in,
                              void* d_out, int out_size, void* d_ws, size_t ws_size,
                              hipStream_t stream) {
  (void)n_in; (void)out_size; (void)ws_size;
  const int IN_DIM = 128, HID = 256, SIGD = 64, H = 8, L = 3;

  // Inputs flattened in setup_inputs() insertion order.
  const float* x      = (const float*)d_in[0];   // [N,128]
  const int*   ei     = (const int*)  d_in[1];   // [2,E]
  const float* esig   = (const float*)d_in[2];   // [E,64]
  const float* inp_W  = (const float*)d_in[3];
  const float* inp_b  = (const float*)d_in[4];
  const float* sc_W1  = (const float*)d_in[5];
  const float* sc_b1  = (const float*)d_in[6];
  const float* sc_W2  = (const float*)d_in[7];
  const float* sc_b2  = (const float*)d_in[8];
  const float* sc_g   = (const float*)d_in[9];
  const float* sc_be  = (const float*)d_in[10];
  const float* fin_g  = (const float*)d_in[11];
  const float* fin_be = (const float*)d_in[12];

  const int Nn = in_sizes[0] / IN_DIM;   // 10000
  const int Ee = in_sizes[1] / 2;        // 320000

  // ---- workspace carve (~708 MB total) ----
  char* ws = (char*)d_ws;
  size_t off = 0;
  auto carve = [&](size_t bytes) -> void* {
    void* p = ws + off;
    off = (off + bytes + 255) & ~(size_t)255;
    return p;
  };
  float*    sig   = (float*)   carve((size_t)Ee * HID * 4);
  float*    eproj = (float*)   carve((size_t)Ee * HID * 4);  // tmp1 aliases here
  float*    h     = (float*)   carve((size_t)Nn * HID * 4);
  float*    xl    = (float*)   carve((size_t)Nn * HID * 4);
  float*    xr    = (float*)   carve((size_t)Nn * HID * 4);
  float*    agg   = (float*)   carve((size_t)Nn * HID * 4);
  float*    lg    = (float*)   carve((size_t)Ee * H * 4);
  unsigned* nmax  = (unsigned*)carve((size_t)Nn * H * 4);
  float*    nsum  = (float*)   carve((size_t)Nn * H * 4);

  auto gemm = [&](const float* A, const float* W, const float* b, float* C,
                  int M, int K, int Ncols) {
    int blocks = ((M + BM - 1) / BM) * (Ncols / BN);
    gemm_wmma_f32_kernel<<<blocks, 256, 0, stream>>>(A, W, b, C, M, K, Ncols);
  };

  // ---- signature compressor: LN(GELU(esig@W1+b1)@W2+b2) ----
  float* tmp1 = eproj;                                   // [E,128] scratch
  gemm(esig, sc_W1, sc_b1, tmp1, Ee, SIGD, 128);
  int ng = Ee * 128;
  gelu_kernel<<<(ng + 255) / 256, 256, 0, stream>>>(tmp1, ng);
  gemm(tmp1, sc_W2, sc_b2, sig, Ee, 128, HID);
  resid_ln_kernel<<<(Ee + 7) / 8, 256, 0, stream>>>(sig, nullptr, nullptr,
                                                    sc_g, sc_be, sig, Ee);

  // ---- input projection ----
  gemm(x, inp_W, inp_b, h, Nn, IN_DIM, HID);

  // ---- GATv2 layers ----
  for (int l = 0; l < L; ++l) {
    const float* Wl   = (const float*)d_in[13 + 9 * l + 0];
    const float* bl   = (const float*)d_in[13 + 9 * l + 1];
    const float* Wr   = (const float*)d_in[13 + 9 * l + 2];
    const float* br   = (const float*)d_in[13 + 9 * l + 3];
    const float* We   = (const float*)d_in[13 + 9 * l + 4];
    const float* att  = (const float*)d_in[13 + 9 * l + 5];
    const float* bias = (const float*)d_in[13 + 9 * l + 6];
    const float* ln_g = (const float*)d_in[13 + 9 * l + 7];
    const float* ln_b = (const float*)d_in[13 + 9 * l + 8];

    gemm(h,   Wl, bl,      xl,    Nn, HID, HID);
    gemm(h,   Wr, br,      xr,    Nn, HID, HID);
    gemm(sig, We, nullptr, eproj, Ee, HID, HID);   // dominant GEMM (42 GFLOP)

    int nh = Nn * H, na = Nn * HID;
    fill_u32_kernel<<<(nh + 255) / 256, 256, 0, stream>>>(nmax, ORD_NEG_INF, nh);
    fill_u32_kernel<<<(nh + 255) / 256, 256, 0, stream>>>((unsigned*)nsum, 0u, nh);
    fill_u32_kernel<<<(na + 255) / 256, 256, 0, stream>>>((unsigned*)agg,  0u, na);

    edge_logits_kernel<<<(Ee + 7) / 8, 256, 0, stream>>>(xl, xr, eproj, ei,
                                                         att, lg, nmax, Ee);
    edge_exp_kernel<<<(Ee * H + 255) / 256, 256, 0, stream>>>(ei, nmax, lg,
                                                              nsum, Ee);
    edge_scatter_kernel<<<(Ee + 7) / 8, 256, 0, stream>>>(xl, ei, lg, nsum,
                                                          agg, Ee);
    resid_ln_kernel<<<(Nn + 7) / 8, 256, 0, stream>>>(h, agg, bias,
                                                      ln_g, ln_b, h, Nn);
  }

  // ---- final LayerNorm -> output ----
  resid_ln_kernel<<<(Nn + 7) / 8, 256, 0, stream>>>(h, nullptr, nullptr,
                                                    fin_g, fin_be,
                                                    (float*)d_out, Nn);
}